// SpatialMemory_32667521253865
// MI455X (gfx1250) — compile-verified
//
#include <hip/hip_runtime.h>

// SpatialMemory: EMA scatter + gather + place-cell Gaussians, reformulated as
// masked f32 WMMA GEMMs for gfx1250.
//
// Inputs (setup_inputs order):
//  0 position (4096,2) f32 | 1 features (4096,256) f32 | 2 spatial_map (512,512,256) f32
//  3 visit_count (512,512) f32 | 4 place_cell_centers (100,2) f32 | 5 place_cell_widths (100,) f32
// Output: (4096, 357) f32 = [feats 256 | novelty 1 | act 100]

#define MAP_SZ   512
#define FEATS    256
#define BATCH    4096
#define N_PLACE  100
#define OUT_COLS 357
#define ALPHA    0.1f

typedef float v2f __attribute__((ext_vector_type(2)));
typedef float v8f __attribute__((ext_vector_type(8)));

// ---- kernel 1: per-element grid cell id -------------------------------------
__global__ void k_cells(const float* __restrict__ pos, int* __restrict__ cells) {
  int b = blockIdx.x * blockDim.x + threadIdx.x;
  if (b >= BATCH) return;
  float px = pos[2 * b + 0], py = pos[2 * b + 1];
  int gx = (int)(px * (float)MAP_SZ);          // trunc-toward-zero like astype(int32)
  int gy = (int)(py * (float)MAP_SZ);
  gx = gx < 0 ? 0 : (gx > MAP_SZ - 1 ? MAP_SZ - 1 : gx);
  gy = gy < 0 ? 0 : (gy > MAP_SZ - 1 ? MAP_SZ - 1 : gy);
  cells[b] = gx * MAP_SZ + gy;
}

// ---- kernel 2: EMA rank weights, initial-map decay, novelty -----------------
__global__ void k_weights(const int* __restrict__ cells,
                          const float* __restrict__ vcount0,
                          float* __restrict__ w, float* __restrict__ decay,
                          float* __restrict__ out) {
  int b = blockIdx.x * blockDim.x + threadIdx.x;
  if (b >= BATCH) return;
  int c = cells[b];
  int later = 0, total = 0;
  const int4* c4 = (const int4*)cells;
  for (int j4 = 0; j4 < BATCH / 4; ++j4) {
    int4 v = c4[j4];
    int j = j4 * 4;
    int m0 = (v.x == c), m1 = (v.y == c), m2 = (v.z == c), m3 = (v.w == c);
    total += m0 + m1 + m2 + m3;
    later += (m0 & (j + 0 > b)) + (m1 & (j + 1 > b)) +
             (m2 & (j + 2 > b)) + (m3 & (j + 3 > b));
  }
  w[b]     = ALPHA * __powf(1.0f - ALPHA, (float)later);  // 0.1 * 0.9^rank
  decay[b] = __powf(1.0f - ALPHA, (float)total);          // initial-map survival
  float vc_final = vcount0[c] + (float)total;
  out[(size_t)b * OUT_COLS + FEATS] = __expf(-vc_final * 0.1f);  // novelty
}

// ---- kernel 3: feats = decay*smap0 + M @ features  (masked f32 WMMA GEMM) ---
// Block = 128 threads (4 waves) covering 16 batch rows x 256 feature cols;
// each wave owns 4 adjacent 16x16 output tiles (64 cols) and scans K=BATCH.
// cells/w staged to LDS via gfx1250 async-to-LDS; scan checks 32 K per step.
__global__ __launch_bounds__(128)
void k_feats(const int* __restrict__ cells, const float* __restrict__ w,
             const float* __restrict__ decay,
             const float* __restrict__ feat, const float* __restrict__ smap0,
             float* __restrict__ out) {
  __shared__ int   sh_cells[BATCH];   // 16 KB
  __shared__ float sh_w[BATCH];       // 16 KB

  const int tid = threadIdx.x;

  // ---- async stage: 2 x 16 KB global -> LDS (b128, tracked by ASYNCcnt) ----
  for (int i = 0; i < BATCH / 4; i += 128) {
    int idx = (i + tid) * 4;                       // dword index, 16B aligned
    unsigned ldsc = (unsigned)(unsigned long long)&sh_cells[idx];
    unsigned ldsw = (unsigned)(unsigned long long)&sh_w[idx];
    unsigned voff = (unsigned)idx * 4u;            // byte offset
    asm volatile("global_load_async_to_lds_b128 %0, %1, %2 offset:0"
                 :: "v"(ldsc), "v"(voff), "s"(cells) : "memory");
    asm volatile("global_load_async_to_lds_b128 %0, %1, %2 offset:0"
                 :: "v"(ldsw), "v"(voff), "s"(w) : "memory");
  }
  asm volatile("s_wait_asynccnt 0" ::: "memory");
  __syncthreads();

  const int lane    = tid & 31;
  const int wave    = tid >> 5;
  const int rowbase = blockIdx.x * 16;
  const int colbase = wave * 64;
  const int m    = lane & 15;   // row (A) / col (B,D) within tile
  const int kg   = lane >> 4;   // 0 -> K{0,1}, 1 -> K{2,3}
  const int rsel = kg;          // D: lanes 16-31 hold M = 8..15

  const int rowcell = sh_cells[rowbase + m];
  // Broadcast all 16 row cell-ids to every lane: full 16-row x 32-K coverage
  // with a single b32 load per lane per scan step.
  int rc[16];
#pragma unroll
  for (int i = 0; i < 16; ++i) rc[i] = __shfl(rowcell, i);

  // Accumulators seeded with the decayed initial spatial_map term.
  v8f acc[4];
#pragma unroll
  for (int t = 0; t < 4; ++t) {
#pragma unroll
    for (int r = 0; r < 8; ++r) {
      int row = rowbase + rsel * 8 + r;
      size_t base = (size_t)sh_cells[row] * FEATS;
      acc[t][r] = decay[row] * smap0[base + colbase + t * 16 + m];
    }
  }

  for (int kb = 0; kb < BATCH; kb += 32) {
    int c = sh_cells[kb + lane];
    int any = 0;
#pragma unroll
    for (int i = 0; i < 16; ++i) any |= (c == rc[i]);
    unsigned mask = (unsigned)__ballot(any);       // wave-uniform hit mask
    while (mask) {                                 // uniform loop: EXEC stays full
      int bit = __builtin_ctz(mask);
      int blk = bit >> 2;                          // 4-wide K block containing hit
      mask &= ~(15u << (blk * 4));
      int k0 = kb + blk * 4 + 2 * kg;
      int c0 = sh_cells[k0], c1 = sh_cells[k0 + 1];
      float wv0 = sh_w[k0], wv1 = sh_w[k0 + 1];    // unconditional LDS loads
      v2f a;
      a.x = (c0 == rowcell) ? wv0 : 0.0f;          // arithmetic select, no exec mask
      a.y = (c1 == rowcell) ? wv1 : 0.0f;
      const float* f0 = feat + (size_t)k0 * FEATS;
      const float* f1 = f0 + FEATS;
#pragma unroll
      for (int t = 0; t < 4; ++t) {
        int col = colbase + t * 16 + m;
        v2f bt;
        bt.x = f0[col];
        bt.y = f1[col];
        acc[t] = __builtin_amdgcn_wmma_f32_16x16x4_f32(
            false, a, false, bt, (short)0, acc[t], false, false);
      }
    }
  }

#pragma unroll
  for (int r = 0; r < 8; ++r) {
    int row = rowbase + rsel * 8 + r;
#pragma unroll
    for (int t = 0; t < 4; ++t)
      out[(size_t)row * OUT_COLS + colbase + t * 16 + m] = acc[t][r];
  }
}

// ---- kernel 4: place-cell activations via exact K=4 WMMA --------------------
// d2[b,j] = [px,py,|p|^2,1] . [-2cx,-2cy,1,|c|^2]  ==  ||p-c||^2
__global__ __launch_bounds__(128)
void k_act(const float* __restrict__ pos, const float* __restrict__ centers,
           const float* __restrict__ widths, float* __restrict__ out) {
  const int lane    = threadIdx.x & 31;
  const int wave    = threadIdx.x >> 5;
  const int rowbase = (blockIdx.x * 4 + wave) * 16;
  const int m    = lane & 15;
  const int kg   = lane >> 4;
  const int rsel = kg;

  int row = rowbase + m;
  float px = pos[2 * row + 0], py = pos[2 * row + 1];
  v2f a;
  a.x = kg ? (px * px + py * py) : px;   // K2=|p|^2 : K0=px
  a.y = kg ? 1.0f                : py;   // K3=1     : K1=py

  for (int ct = 0; ct < 7; ++ct) {       // 7*16 = 112 >= 100 place cells
    int col = ct * 16 + m;
    int cc  = col < N_PLACE ? col : N_PLACE - 1;   // clamp, avoid divergent loads
    float cx = centers[2 * cc + 0], cy = centers[2 * cc + 1];
    v2f bv;
    bv.x = kg ? 1.0f                : -2.0f * cx;  // K2 : K0
    bv.y = kg ? (cx * cx + cy * cy) : -2.0f * cy;  // K3 : K1
    v8f d = {};
    d = __builtin_amdgcn_wmma_f32_16x16x4_f32(
        false, a, false, bv, (short)0, d, false, false);
    float wd  = widths[cc];
    float inv = 0.5f / (wd * wd);                  // 1/(2w^2)
    if (col < N_PLACE) {
#pragma unroll
      for (int r = 0; r < 8; ++r) {
        int orow = rowbase + rsel * 8 + r;
        out[(size_t)orow * OUT_COLS + FEATS + 1 + col] = __expf(-d[r] * inv);
      }
    }
  }
}

// ---- launch -----------------------------------------------------------------
extern "C" void kernel_launch(void* const* d_in, const int* in_sizes, int n_in,
                              void* d_out, int out_size, void* d_ws, size_t ws_size,
                              hipStream_t stream) {
  const float* pos   = (const float*)d_in[0];
  const float* feat  = (const float*)d_in[1];
  const float* smap0 = (const float*)d_in[2];
  const float* vc0   = (const float*)d_in[3];
  const float* ctr   = (const float*)d_in[4];
  const float* wid   = (const float*)d_in[5];
  float* out = (float*)d_out;

  int*   cells = (int*)d_ws;                                      // 16 KB
  float* w     = (float*)((char*)d_ws + BATCH * sizeof(int));     // 16 KB
  float* decay = (float*)((char*)d_ws + 2 * BATCH * sizeof(int)); // 16 KB

  k_cells  <<<BATCH / 256, 256, 0, stream>>>(pos, cells);
  k_weights<<<BATCH / 256, 256, 0, stream>>>(cells, vc0, w, decay, out);
  k_feats  <<<BATCH / 16, 128, 0, stream>>>(cells, w, decay, feat, smap0, out);
  k_act    <<<BATCH / 64, 128, 0, stream>>>(pos, ctr, wid, out);
}